// SpaceEdgeConv_25503515804301
// MI455X (gfx1250) — compile-verified
//
#include <hip/hip_runtime.h>
#include <stdint.h>

// Problem constants (from reference): N=256 batches, P=256 points, F=128 feats,
// OUT=128, n_dim=2 (coords are x[:, :, 126:128]).
constexpr int kN   = 256;
constexpr int kP   = 256;
constexpr int kF   = 128;
constexpr int kOut = 128;
constexpr int kRow = 129;   // LDS row stride in dwords: 128 data + 1 pad (TDM pad_enable)

typedef float v8f __attribute__((ext_vector_type(8)));
typedef float v2f __attribute__((ext_vector_type(2)));
typedef unsigned int u32x4 __attribute__((ext_vector_type(4)));
typedef int i32x4 __attribute__((ext_vector_type(4)));
typedef int i32x8 __attribute__((ext_vector_type(8)));

// ---------------------------------------------------------------------------
// TDM: DMA one batch's x-slab (256 rows x 128 f32 = 128KB) global -> LDS.
// D# per CDNA5 ISA. 2D tile, data_size=4B,
// pad_enable=1 / pad_interval=128 dwords / pad_amount=1 dword so each
// 128-dword row lands at LDS stride 129 dwords (bank-conflict-free WMMA
// column reads).
// ---------------------------------------------------------------------------
__device__ __forceinline__ void tdm_load_x_to_lds(const float* gx, unsigned lds_off) {
  unsigned long long ga = (unsigned long long)(uintptr_t)gx;
  u32x4 g0;
  g0[0] = 1u;                                              // count=1, user descriptor
  g0[1] = lds_off;                                         // lds_addr (bytes)
  g0[2] = (unsigned)(ga & 0xFFFFFFFFull);                  // global_addr[31:0]
  g0[3] = (unsigned)((ga >> 32) & 0x1FFFFFFull) | (2u << 30); // global_addr[56:32] | type=2
  i32x8 g1;
  g1[0] = (int)((2u << 16) | (1u << 20) | (6u << 22));     // data_size=4B, pad_en, pad_interval=128dw, pad_amount=1dw
  g1[1] = (int)(128u << 16);                               // tensor_dim0 = 128 (lo16)
  g1[2] = (int)(256u << 16);                               // tensor_dim0 hi=0 | tensor_dim1 = 256 (lo16)
  g1[3] = (int)(128u << 16);                               // tensor_dim1 hi=0 | tile_dim0 = 128
  g1[4] = 256;                                             // tile_dim1 = 256, tile_dim2 = 0
  g1[5] = 128;                                             // tensor_dim0_stride = 128 (lo32)
  g1[6] = (int)(0x8000u << 16);                            // dim0_stride hi=0 | tensor_dim1_stride lo16 = 32768
  g1[7] = 0;                                               // tensor_dim1_stride hi
  i32x4 z4 = {0, 0, 0, 0};
#if __has_include(<hip/amd_detail/amd_gfx1250_TDM.h>)
  i32x8 z8 = {0, 0, 0, 0, 0, 0, 0, 0};
  __builtin_amdgcn_tensor_load_to_lds(g0, g1, z4, z4, z8, 0);   // clang-23 / therock: 6 args
#else
  __builtin_amdgcn_tensor_load_to_lds(g0, g1, z4, z4, 0);       // ROCm 7.2 / clang-22: 5 args
#endif
}

// ---------------------------------------------------------------------------
// Fused SpaceEdgeConv kernel: one workgroup per batch n, 256 threads = 8 waves.
// ---------------------------------------------------------------------------
__global__ __launch_bounds__(256) void SpaceEdgeConv_fused(
    const float* __restrict__ x, const float* __restrict__ mask,
    const float* __restrict__ A, const float* __restrict__ bias,
    float* __restrict__ out) {
  extern __shared__ float smem[];
  float* xs  = smem;                      // [256][129] f32: x slab (TDM-loaded)
  float* ssm = smem + kP * kRow;          // [256][129] f32: ss = (W@mx - x)/norm
  float* msk = smem + 2 * kP * kRow;      // [256] f32

  const int n    = blockIdx.x;
  const int tid  = threadIdx.x;
  const int lane = tid & 31;
  const int wave = tid >> 5;
  const int l15  = lane & 15;
  const int half = (lane >> 4) & 1;       // K-pair select: 0 -> {0,1}, 1 -> {2,3}

  const float* xn = x + (size_t)n * kP * kF;

  // ---- Phase 1: TDM x -> LDS, mask -> LDS, prefetch A into L2 ----
  if (wave == 0) {
    tdm_load_x_to_lds(xn, (unsigned)(uintptr_t)xs);
  }
  msk[tid] = mask[(size_t)n * kP + tid];
  __builtin_prefetch(&A[(size_t)tid * kOut], 0, 3);   // warm A (128KB, L2-resident)
  if (wave == 0) {
    __builtin_amdgcn_s_wait_tensorcnt(0);
  }
  __syncthreads();

  float norm = 0.0f;
#pragma unroll 8
  for (int j = 0; j < kP; ++j) norm += msk[j];
  const float normInv = 1.0f / norm;

  const v8f vzero = {0.f, 0.f, 0.f, 0.f, 0.f, 0.f, 0.f, 0.f};

  // ---- Phase 2: ss_raw = (exp(-10*dist^2) * mask) @ x, per 16-row strip ----
  for (int s = wave; s < 16; s += 8) {
    const int ib   = s * 16;
    const int irow = ib + l15;                 // A-frag: lane row
    const float cix = xs[irow * kRow + (kF - 2)];
    const float ciy = xs[irow * kRow + (kF - 1)];

    v8f acc[8];
#pragma unroll
    for (int t = 0; t < 8; ++t) acc[t] = vzero;

    for (int kb = 0; kb < kP; kb += 4) {
      const int kk = kb + 2 * half;            // this lane's K pair
      // Build W A-fragment on the fly (gaussian edge weight * mask[j])
      const float dx0 = cix - xs[kk * kRow + (kF - 2)];
      const float dy0 = ciy - xs[kk * kRow + (kF - 1)];
      const float dx1 = cix - xs[(kk + 1) * kRow + (kF - 2)];
      const float dy1 = ciy - xs[(kk + 1) * kRow + (kF - 1)];
      v2f a;
      a.x = __expf(-10.0f * (dx0 * dx0 + dy0 * dy0)) * msk[kk];
      a.y = __expf(-10.0f * (dx1 * dx1 + dy1 * dy1)) * msk[kk + 1];
#pragma unroll
      for (int t = 0; t < 8; ++t) {
        const int o = t * 16 + l15;
        v2f b;
        b.x = xs[kk * kRow + o];
        b.y = xs[(kk + 1) * kRow + o];
        acc[t] = __builtin_amdgcn_wmma_f32_16x16x4_f32(
            false, a, false, b, (short)0, acc[t], false, false);
      }
    }
    // Epilogue: ss = (ss_raw - x) / norm  (C layout: vgpr v -> M = v + 8*half)
#pragma unroll
    for (int t = 0; t < 8; ++t) {
#pragma unroll
      for (int v = 0; v < 8; ++v) {
        const int row = ib + v + 8 * half;
        const int col = t * 16 + l15;
        ssm[row * kRow + col] = (acc[t][v] - xs[row * kRow + col]) * normInv;
      }
    }
  }
  __syncthreads();

  // ---- Phase 3: out = [x, ss] @ A + b, masked. Wave w owns A cols [16w,16w+16) ----
  const int obase = wave * 16;
  const float bval = bias[obase + l15];

  for (int pass = 0; pass < 2; ++pass) {       // 8 row-strips per pass (VGPR budget)
    v8f acc[8];
#pragma unroll
    for (int j = 0; j < 8; ++j) acc[j] = vzero;

    for (int kb = 0; kb < 2 * kF; kb += 4) {   // K over concat feature dim = 256
      const int kk = kb + 2 * half;
      v2f b;                                   // B-frag from A (L2-resident)
      b.x = A[(size_t)kk * kOut + obase + l15];
      b.y = A[(size_t)(kk + 1) * kOut + obase + l15];
      const float* fb = (kb < kF) ? xs : ssm;  // uniform per kb: x half vs ss half
      const int kloc  = (kb < kF) ? kk : (kk - kF);
#pragma unroll
      for (int j = 0; j < 8; ++j) {
        const int row = (pass * 8 + j) * 16 + l15;
        v2f a;
        a.x = fb[row * kRow + kloc];
        a.y = fb[row * kRow + kloc + 1];
        acc[j] = __builtin_amdgcn_wmma_f32_16x16x4_f32(
            false, a, false, b, (short)0, acc[j], false, false);
      }
    }
    // Epilogue: + bias, * mask[row], store to HBM
#pragma unroll
    for (int j = 0; j < 8; ++j) {
      const int ib = (pass * 8 + j) * 16;
#pragma unroll
      for (int v = 0; v < 8; ++v) {
        const int row = ib + v + 8 * half;
        const float m = msk[row];
        out[((size_t)n * kP + row) * kOut + obase + l15] = (acc[j][v] + bval) * m;
      }
    }
  }
}

extern "C" void kernel_launch(void* const* d_in, const int* in_sizes, int n_in,
                              void* d_out, int out_size, void* d_ws, size_t ws_size,
                              hipStream_t stream) {
  const float* x    = (const float*)d_in[0];
  const float* mask = (const float*)d_in[1];
  const float* A    = (const float*)d_in[2];
  const float* b    = (const float*)d_in[3];
  // d_in[4] = n_dim (always 2 in this workload)
  float* out = (float*)d_out;

  const size_t smemBytes = (size_t)(2 * kP * kRow + kP) * sizeof(float); // ~259 KB of 320 KB WGP LDS
  hipFuncSetAttribute(reinterpret_cast<const void*>(SpaceEdgeConv_fused),
                      hipFuncAttributeMaxDynamicSharedMemorySize, (int)smemBytes);

  SpaceEdgeConv_fused<<<dim3(kN), dim3(256), smemBytes, stream>>>(x, mask, A, b, out);
}